// GNN_37623913513027
// MI455X (gfx1250) — compile-verified
//
#include <hip/hip_runtime.h>
#include <hip/hip_bf16.h>

// ---------------- constants matching the reference ----------------
#define S_      64
#define G_      256
#define N_      64
#define E_      256
#define GENES_  20000
#define DH_     64      // HID*HEADS
#define HEADS_  4
#define HID_    16
#define NEDGE_  (E_ + N_)   // edges + self loops = 320
#define LDP     72          // padded LDS row stride (floats): 72 mod 64 -> conflict-free cols

typedef __attribute__((ext_vector_type(2))) float v2f;
typedef __attribute__((ext_vector_type(8))) float v8f;

// Wave-cooperative 64x64x64 f32 GEMM using V_WMMA_F32_16X16X4_F32.
// A: [64][LDP] row-major (MxK).  D: [64][LDP].
// BT=true : B given transposed, B[n][k] (so the B-frag load is a contiguous b64).
// BT=false: B given as B[k][n] (two b32 column reads; stride-72 keeps banks disjoint).
template <bool BT>
__device__ __forceinline__ void wave_gemm64(const float* __restrict__ A,
                                            const float* __restrict__ B,
                                            float* __restrict__ D) {
    const int lane = threadIdx.x & 31;
    const int wave = threadIdx.x >> 5;     // 0..7 (8 waves)
    const int m    = lane & 15;            // M (A) / N (B) index within tile
    const int kh   = lane >> 4;            // K half: lanes 0-15 -> K{0,1}, 16-31 -> K{2,3}
    for (int t = wave; t < 16; t += 8) {   // 4x4 tile grid, 2 tiles per wave
        const int mi = (t >> 2) * 16;
        const int ni = (t & 3) * 16;
        v8f acc = {};
#pragma unroll
        for (int k0 = 0; k0 < 64; k0 += 4) {
            const int ka = k0 + kh * 2;
            v2f a, b;
            a.x = A[(mi + m) * LDP + ka];
            a.y = A[(mi + m) * LDP + ka + 1];
            if (BT) {
                b.x = B[(ni + m) * LDP + ka];
                b.y = B[(ni + m) * LDP + ka + 1];
            } else {
                b.x = B[ka * LDP + (ni + m)];
                b.y = B[(ka + 1) * LDP + (ni + m)];
            }
            // D = A(16x4) * B(4x16) + C, all f32 (exact-precision path)
            acc = __builtin_amdgcn_wmma_f32_16x16x4_f32(
                false, a, false, b, (short)0, acc, false, false);
        }
#pragma unroll
        for (int v = 0; v < 8; ++v)        // C/D layout: row = kh*8+v, col = lane&15
            D[(mi + kh * 8 + v) * LDP + (ni + m)] = acc[v];
    }
}

__device__ __forceinline__ float leaky02(float v) { return v > 0.f ? v : 0.2f * v; }

// One block per graph g; loops over all S samples. 256 threads = 8 waves.
__global__ __launch_bounds__(256)
void gat_stage(const float* __restrict__ gx,       // [S, GENES]
               const float* __restrict__ W1g,      // [64]
               const float* __restrict__ as1,      // [4,16]
               const float* __restrict__ ad1,      // [4,16]
               const float* __restrict__ b1g,      // [64]
               const float* __restrict__ W2g,      // [64,64]
               const float* __restrict__ as2g,     // [64]
               const float* __restrict__ ad2g,     // [64]
               const float* __restrict__ b2g,      // [64]
               const float* __restrict__ fcwg,     // [64]
               const float* __restrict__ fcbg,     // [1]
               const int*   __restrict__ gene_idx, // [G,N]
               const int*   __restrict__ edge_src, // [G,E]
               const int*   __restrict__ edge_dst, // [G,E]
               float* __restrict__ go)             // [S,G] output scalars
{
    __shared__ float W2t[64 * LDP];   // W2 transposed: W2t[n][k] = W2[k][n]
    __shared__ float bufA[64 * LDP];  // h1, later out2
    __shared__ float lin2[64 * LDP];  // h1 @ W2
    __shared__ float Aat[64 * LDP];   // dense attention matrix (layer 2)
    __shared__ float xsh[64];
    __shared__ float s2[64], d2[64];
    __shared__ float W1s[64], b1s[64], b2s[64], as2[64], ad2[64], fcw[64];
    __shared__ float cs1[4], cd1[4];
    __shared__ float red[64];
    __shared__ int gidx[64];
    __shared__ int esrc[NEDGE_];      // CSR (by dst) source list
    __shared__ int offs[65];
    __shared__ int cur[64];

    const int g = blockIdx.x;
    const int t = threadIdx.x;

    // ---- per-graph setup: indices, CSR-by-dst (edges + self loops), weights ----
    if (t < 64) { gidx[t] = gene_idx[g * N_ + t]; cur[t] = 0; }
    __syncthreads();
    if (t < E_) atomicAdd(&cur[edge_dst[g * E_ + t]], 1);
    if (t < N_) atomicAdd(&cur[t], 1);               // self loop
    __syncthreads();
    if (t == 0) {
        int acc = 0;
        for (int n = 0; n < 64; ++n) { offs[n] = acc; acc += cur[n]; }
        offs[64] = acc;                               // == 320
    }
    __syncthreads();
    if (t < 64) cur[t] = offs[t];
    __syncthreads();
    if (t < E_) {
        int d = edge_dst[g * E_ + t];
        int p = atomicAdd(&cur[d], 1);
        esrc[p] = edge_src[g * E_ + t];
    }
    if (t < N_) { int p = atomicAdd(&cur[t], 1); esrc[p] = t; }
    if (t < 64) {
        W1s[t] = W1g[t]; b1s[t] = b1g[t]; b2s[t] = b2g[t];
        as2[t] = as2g[t]; ad2[t] = ad2g[t]; fcw[t] = fcwg[t];
    }
    for (int i = t; i < 64 * 64; i += 256) {          // W2 transpose into LDS
        int k = i >> 6, n = i & 63;
        W2t[n * LDP + k] = W2g[i];
    }
    if (t < HEADS_) {                                  // layer-1 attention collapses to scalars
        float a = 0.f, b = 0.f;
        for (int c = 0; c < HID_; ++c) {
            a += W1g[t * HID_ + c] * as1[t * HID_ + c];
            b += W1g[t * HID_ + c] * ad1[t * HID_ + c];
        }
        cs1[t] = a; cd1[t] = b;
    }
    __syncthreads();

    // ---------------------------- sample loop ----------------------------
    for (int s = 0; s < S_; ++s) {
        if (t < 64) xsh[t] = gx[(size_t)s * GENES_ + gidx[t]];
        __syncthreads();

        // GAT layer 1: per (node, head) scalar softmax -> h1 row (relu)
        {
            const int n = t & 63, h = t >> 6;          // 64 nodes x 4 heads
            const float xd = xsh[n];
            const int lo = offs[n], hi = offs[n + 1];
            float mx = -1e30f;
            for (int j = lo; j < hi; ++j)
                mx = fmaxf(mx, leaky02(cs1[h] * xsh[esrc[j]] + cd1[h] * xd));
            float den = 0.f, agg = 0.f;
            for (int j = lo; j < hi; ++j) {
                float xs = xsh[esrc[j]];
                float p  = __expf(leaky02(cs1[h] * xs + cd1[h] * xd) - mx);
                den += p; agg += p * xs;
            }
            agg /= den;
#pragma unroll
            for (int c = 0; c < HID_; ++c) {
                int hc = h * HID_ + c;
                float v = fmaf(W1s[hc], agg, b1s[hc]);
                bufA[n * LDP + hc] = v > 0.f ? v : 0.f;
            }
        }
        __syncthreads();

        // lin2 = h1 @ W2  (f32 WMMA, B transposed in LDS)
        wave_gemm64<true>(bufA, W2t, lin2);
        __syncthreads();

        // attention-2 node scalars + clear dense attention matrix
        if (t < 64) {
            float a = 0.f;
            for (int k = 0; k < 64; ++k) a += lin2[t * LDP + k] * as2[k];
            s2[t] = a;
        } else if (t < 128) {
            const int n = t - 64;
            float a = 0.f;
            for (int k = 0; k < 64; ++k) a += lin2[n * LDP + k] * ad2[k];
            d2[n] = a;
        }
        for (int i = t; i < 64 * LDP; i += 256) Aat[i] = 0.f;
        __syncthreads();

        // softmax over incoming edges; scatter alpha into dense A (row owned by one thread)
        if (t < 64) {
            const int n = t, lo = offs[n], hi = offs[n + 1];
            const float dn = d2[n];
            float mx = -1e30f;
            for (int j = lo; j < hi; ++j)
                mx = fmaxf(mx, leaky02(s2[esrc[j]] + dn));
            float den = 0.f;
            for (int j = lo; j < hi; ++j)
                den += __expf(leaky02(s2[esrc[j]] + dn) - mx);
            const float inv = 1.f / den;
            for (int j = lo; j < hi; ++j)
                Aat[n * LDP + esrc[j]] += __expf(leaky02(s2[esrc[j]] + dn) - mx) * inv;
        }
        __syncthreads();

        // out2 = A_dense @ lin2  (f32 WMMA, B row-major)
        wave_gemm64<false>(Aat, lin2, bufA);
        __syncthreads();

        // mean-pool over nodes (relu(+b2)) then dot with fc_w
        if (t < 64) {
            float pk = 0.f;
            for (int n = 0; n < 64; ++n) {
                float v = bufA[n * LDP + t] + b2s[t];
                pk += v > 0.f ? v : 0.f;
            }
            red[t] = (pk * (1.0f / 64.0f)) * fcw[t];
        }
        __syncthreads();
        if (t == 0) {
            float sm = 0.f;
            for (int k = 0; k < 64; ++k) sm += red[k];
            go[s * G_ + g] = sm + fcbg[0];
        }
        __syncthreads();   // protect red/xsh/buffers for next sample
    }
}

// Tiny 256->128->128->1 MLP over the S=64 samples. One block, 128 threads.
__global__ __launch_bounds__(128)
void mlp_stage(const float* __restrict__ go,     // [S,G]
               const float* __restrict__ fc1w, const float* __restrict__ fc1b,
               const float* __restrict__ fc2w, const float* __restrict__ fc2b,
               const float* __restrict__ fc3w, const float* __restrict__ fc3b,
               float* __restrict__ out)          // [S]
{
    __shared__ float h1s[S_ * 128];
    __shared__ float h2s[S_ * 128];
    const int j = threadIdx.x;
    for (int s = 0; s < S_; ++s) {
        float acc = fc1b[j];
        for (int k = 0; k < G_; ++k)
            acc = fmaf(go[s * G_ + k], fc1w[k * 128 + j], acc);
        h1s[s * 128 + j] = acc > 0.f ? acc : 0.f;
    }
    __syncthreads();
    for (int s = 0; s < S_; ++s) {
        float acc = fc2b[j];
        for (int k = 0; k < 128; ++k)
            acc = fmaf(h1s[s * 128 + k], fc2w[k * 128 + j], acc);
        h2s[s * 128 + j] = acc > 0.f ? acc : 0.f;
    }
    __syncthreads();
    if (j < S_) {
        float acc = fc3b[0];
        for (int k = 0; k < 128; ++k)
            acc = fmaf(h2s[j * 128 + k], fc3w[k], acc);
        out[j] = acc;
    }
}

extern "C" void kernel_launch(void* const* d_in, const int* in_sizes, int n_in,
                              void* d_out, int out_size, void* d_ws, size_t ws_size,
                              hipStream_t stream) {
    const float* gx    = (const float*)d_in[0];
    const float* W1    = (const float*)d_in[1];
    const float* as1   = (const float*)d_in[2];
    const float* ad1   = (const float*)d_in[3];
    const float* b1    = (const float*)d_in[4];
    const float* W2    = (const float*)d_in[5];
    const float* as2   = (const float*)d_in[6];
    const float* ad2   = (const float*)d_in[7];
    const float* b2    = (const float*)d_in[8];
    const float* fcw   = (const float*)d_in[9];
    const float* fcb   = (const float*)d_in[10];
    const float* fc1w  = (const float*)d_in[11];
    const float* fc1b  = (const float*)d_in[12];
    const float* fc2w  = (const float*)d_in[13];
    const float* fc2b  = (const float*)d_in[14];
    const float* fc3w  = (const float*)d_in[15];
    const float* fc3b  = (const float*)d_in[16];
    const int*   gidx  = (const int*)d_in[17];
    const int*   esrc  = (const int*)d_in[18];
    const int*   edst  = (const int*)d_in[19];

    float* go = (float*)d_ws;                      // [S, G] scratch (64 KB)

    gat_stage<<<G_, 256, 0, stream>>>(gx, W1, as1, ad1, b1, W2, as2, ad2, b2,
                                      fcw, fcb, gidx, esrc, edst, go);
    mlp_stage<<<1, 128, 0, stream>>>(go, fc1w, fc1b, fc2w, fc2b, fc3w, fc3b,
                                     (float*)d_out);
}